// SGO_Loss_Prod_6751688589549
// MI455X (gfx1250) — compile-verified
//
#include <hip/hip_runtime.h>
#include <math.h>

// ---------------------------------------------------------------------------
// SGO symmetry-operation loss for MI455X (gfx1250, wave32).
//
// dist^2(i,j) for 16x16 pair tiles via V_WMMA_F32_16X16X4_F32:
//   A (16x4)  = [-2fx, -2fy, -2fz, |f|^2]  per dst atom row  (loop-invariant)
//   B (4x16)  = [gx; gy; gz; 1]            per src (shifted) atom column
//   C         = inline 0
//   D = A*B   = |f|^2 - 2 f.g      mask:  D <= r^2 - |g|^2   (per-column thr)
// thr is derived from the stored |raw|^2 so the whole atom record is one
// ds_load_b128:  thr = (R2 - |s|^2 - raw.w) - 2*raw.s
// Masked per-component pair sums use the algebraic split
//   sum m (f-g)^2 = sum_r f^2*cnt - 2 sum_r f*sum(m g) + sum m g^2
// so everything stays in the native WMMA C/D register layout.
// j-tiles are software-pipelined (prefetch t+1 / WMMA t / mask t-1); the
// shift decode is hoisted per 16-tile group and the tile loop fully unrolled
// so LDS loads use immediate offsets.
// ---------------------------------------------------------------------------

typedef __attribute__((ext_vector_type(2))) float v2f;
typedef __attribute__((ext_vector_type(8))) float v8f;

#define B_STRUCT 16
#define N_ATOMS  256
#define K_OPS    16
#define NCFG_PER 17                      // 1 base + 16 ops
#define NCFG     (B_STRUCT * NCFG_PER)   // 272
#define R2       0.16f                   // (R_MAX=0.4)^2, POWER=0

struct GTile { float gx, gy, gz, gx2, gy2, gz2, thr; };

__global__ __launch_bounds__(512)
void sgo_pair_kernel(const float* __restrict__ fracs,
                     const float* __restrict__ oprss,
                     float* __restrict__ S /* [NCFG][3] */) {
    __shared__ float4 f1s[N_ATOMS];      // (x, y, z, |f|^2) per atom
    __shared__ float  Sp[16][3];         // per-wave partial sums

    const int cfg = blockIdx.x;
    const int b   = cfg / NCFG_PER;
    const int k   = cfg % NCFG_PER;      // 0 = base, 1..16 = op index k-1
    const int tid = threadIdx.x;

    // ---- setup: transform fractional coords for this config ----
    if (tid < N_ATOMS) {
        const float* fp = fracs + (size_t)(b * N_ATOMS + tid) * 3;
        float x = fp[0], y = fp[1], z = fp[2];
        if (k > 0) {
            const float* op = oprss + (size_t)(b * K_OPS + (k - 1)) * 9;
            // f1 = f @ opr.T  -> f1_c = dot(opr row c, f)
            float nx = op[0] * x + op[1] * y + op[2] * z;
            float ny = op[3] * x + op[4] * y + op[5] * z;
            float nz = op[6] * x + op[7] * y + op[8] * z;
            x = nx - floorf(nx);         // python-style  % 1.0
            y = ny - floorf(ny);
            z = nz - floorf(nz);
        }
        f1s[tid] = make_float4(x, y, z, x * x + y * y + z * z);
    }
    __syncthreads();

    const int  wave = tid >> 5;          // 0..15 : one 16-row i-tile per wave
    const int  lane = tid & 31;
    const int  ln16 = lane & 15;
    const bool hi   = (lane >= 16);
    const int  i0   = wave * 16;

    // ---- A operand, loop-invariant (ISA 16x4 f32 A-layout, K = v + 2*half):
    // lane<16 : K0=-2fx, K1=-2fy ; lane>=16 : K2=-2fz, K3=|f|^2
    const float4 fa = f1s[i0 + ln16];
    v2f A;
    A.x = hi ? (-2.f * fa.z) : (-2.f * fa.x);
    A.y = hi ? fa.w          : (-2.f * fa.y);

    const v8f Zc = {};                   // C = inline 0

    // per-row (v -> row i0 + v + 8*hi) accumulators + per-lane g^2 sums
    float cnt[8], agx[8], agy[8], agz[8];
#pragma unroll
    for (int v = 0; v < 8; ++v) { cnt[v] = 0.f; agx[v] = 0.f; agy[v] = 0.f; agz[v] = 0.f; }
    float s2x = 0.f, s2y = 0.f, s2z = 0.f;

    auto domm = [&](const GTile& g) -> v8f {
        v2f Bv;
        Bv.x = hi ? g.gz : g.gx;
        Bv.y = hi ? 1.f  : g.gy;
        return __builtin_amdgcn_wmma_f32_16x16x4_f32(
            false, A, false, Bv, (short)0, Zc, false, false);
    };

    auto accum = [&](const v8f& D, const GTile& g) {
        float sm = 0.f;
#pragma unroll
        for (int v = 0; v < 8; ++v) {
            const float m = (D[v] <= g.thr) ? 1.0f : 0.0f;
            cnt[v] += m;
            agx[v] = fmaf(m, g.gx, agx[v]);
            agy[v] = fmaf(m, g.gy, agy[v]);
            agz[v] = fmaf(m, g.gz, agz[v]);
            sm += m;
        }
        s2x = fmaf(g.gx2, sm, s2x);
        s2y = fmaf(g.gy2, sm, s2y);
        s2z = fmaf(g.gz2, sm, s2z);
    };

    // ---- software-pipelined sweep: 27 shifts x 16 column tiles ----
    float4 raw = f1s[ln16];                          // raw coords, tile 0
    GTile gp;
    v8f   Dp;
    {   // prologue: tile 0 (shift = 0 -> thr = R2 - raw.w exactly)
        gp.gx = raw.x; gp.gy = raw.y; gp.gz = raw.z;
        gp.gx2 = gp.gx * gp.gx; gp.gy2 = gp.gy * gp.gy; gp.gz2 = gp.gz * gp.gz;
        gp.thr = R2 - raw.w;
        raw = f1s[16 + ln16];                        // prefetch tile 1
        Dp  = domm(gp);
    }
    // s = 0 (zero shift), tiles 1..15
#pragma unroll
    for (int tt = 1; tt < 16; ++tt) {
        GTile gc;
        gc.gx = raw.x; gc.gy = raw.y; gc.gz = raw.z;
        gc.gx2 = gc.gx * gc.gx; gc.gy2 = gc.gy * gc.gy; gc.gz2 = gc.gz * gc.gz;
        gc.thr = R2 - raw.w;
        raw = f1s[(((tt + 1) & 15) << 4) + ln16];    // prefetch next (wraps)
        const v8f Dc = domm(gc);
        accum(Dp, gp);
        Dp = Dc; gp = gc;
    }
    // s = 1..26
    for (int s = 1; s < 27; ++s) {
        // itertools.product([0,1,-1], repeat=3): digit 0->0, 1->+1, 2->-1
        const int d0 = s / 9, d1 = (s / 3) % 3, d2 = s % 3;
        const float sx = (d0 == 2) ? -1.f : (float)d0;
        const float sy = (d1 == 2) ? -1.f : (float)d1;
        const float sz = (d2 == 2) ? -1.f : (float)d2;
        const float nsx2 = -2.f * sx, nsy2 = -2.f * sy, nsz2 = -2.f * sz;
        const float c0 = R2 - (sx * sx + sy * sy + sz * sz);
#pragma unroll
        for (int tt = 0; tt < 16; ++tt) {
            GTile gc;
            gc.gx = raw.x + sx; gc.gy = raw.y + sy; gc.gz = raw.z + sz;
            gc.gx2 = gc.gx * gc.gx; gc.gy2 = gc.gy * gc.gy; gc.gz2 = gc.gz * gc.gz;
            // thr = R2 - |raw+s|^2 = (c0 - raw.w) - 2*raw.s  (uses all of raw)
            gc.thr = fmaf(raw.x, nsx2,
                     fmaf(raw.y, nsy2,
                     fmaf(raw.z, nsz2, c0 - raw.w)));
            raw = f1s[(((tt + 1) & 15) << 4) + ln16];
            const v8f Dc = domm(gc);
            accum(Dp, gp);
            Dp = Dc; gp = gc;
        }
    }
    accum(Dp, gp);                                   // epilogue

    // ---- finalize:  sum_P (f_c - g_c)^2
    //   = sum_r f_c[r]^2 * cnt  - 2 * sum_r f_c[r]*agc  + sum m*g_c^2
    float px = s2x, py = s2y, pz = s2z;
#pragma unroll
    for (int v = 0; v < 8; ++v) {
        const int r = i0 + v + (hi ? 8 : 0);
        const float4 fr = f1s[r];
        px += fr.x * fr.x * cnt[v] - 2.f * fr.x * agx[v];
        py += fr.y * fr.y * cnt[v] - 2.f * fr.y * agy[v];
        pz += fr.z * fr.z * cnt[v] - 2.f * fr.z * agz[v];
    }
    // deterministic wave32 reduction
#pragma unroll
    for (int off = 16; off > 0; off >>= 1) {
        px += __shfl_xor(px, off, 32);
        py += __shfl_xor(py, off, 32);
        pz += __shfl_xor(pz, off, 32);
    }
    if (lane == 0) { Sp[wave][0] = px; Sp[wave][1] = py; Sp[wave][2] = pz; }
    __syncthreads();

    if (tid < 3) {                        // deterministic serial sum over waves
        float acc = 0.f;
        for (int w = 0; w < 16; ++w) acc += Sp[w][tid];
        S[cfg * 3 + tid] = acc;
    }
}

__global__ void sgo_loss_kernel(const float* __restrict__ S,
                                float* __restrict__ out) {
    if (threadIdx.x == 0 && blockIdx.x == 0) {
        float loss = 0.f;
        for (int b = 0; b < B_STRUCT; ++b) {
            const float* base = S + (size_t)b * NCFG_PER * 3;
            for (int k = 1; k <= K_OPS; ++k) {
                const float* sk = S + (size_t)(b * NCFG_PER + k) * 3;
                const float dx = sk[0] - base[0];
                const float dy = sk[1] - base[1];
                const float dz = sk[2] - base[2];
                loss += sqrtf(dx * dx + dy * dy + dz * dz);
            }
        }
        out[0] = loss * (1.0f / (float)(K_OPS * B_STRUCT));
    }
}

extern "C" void kernel_launch(void* const* d_in, const int* in_sizes, int n_in,
                              void* d_out, int out_size, void* d_ws, size_t ws_size,
                              hipStream_t stream) {
    (void)in_sizes; (void)n_in; (void)out_size; (void)ws_size;
    const float* fracs = (const float*)d_in[0];   // [B*N, 3] f32
    const float* oprss = (const float*)d_in[2];   // [B*K, 3, 3] f32
    float* S = (float*)d_ws;                      // [272][3] partial sums

    sgo_pair_kernel<<<NCFG, 512, 0, stream>>>(fracs, oprss, S);
    sgo_loss_kernel<<<1, 32, 0, stream>>>(S, (float*)d_out);
}